// EdgeConv_30313879175189
// MI455X (gfx1250) — compile-verified
//
#include <hip/hip_runtime.h>

// ---------------------------------------------------------------------------
// EdgeConv message-passing layer for MI455X (gfx1250, wave32).
//   agg   = segment_sum(ef, edge_dst)                       (float atomics)
//   rev_i = sum of ef[j] over all j with key_j == revkey_i  (hash + lists)
//   out   = relu([ef | agg[src]-rev] @ W + b)               (V_WMMA_F32_16X16X4_F32)
// Memory-bound problem (~1.5 GB traffic, 16 GFLOP fp32) -> keep fp32, one
// LDS-staged pass for the GEMM using the fp32 WMMA path.
// ---------------------------------------------------------------------------

typedef float v2f __attribute__((ext_vector_type(2)));
typedef float v8f __attribute__((ext_vector_type(8)));

#define UNITS   64
#define TWOU    128
#define HPAD    132                 // padded LDS row stride (floats): kills bank conflicts
#define EMPTY64 0xFFFFFFFFFFFFFFFFULL

__device__ __forceinline__ unsigned int hash64(unsigned long long k) {
    k ^= k >> 33; k *= 0xff51afd7ed558ccdULL;
    k ^= k >> 33; k *= 0xc4ceb9fe1a85ec53ULL;
    k ^= k >> 33;
    return (unsigned int)k;
}

// ---------------------------------------------------------------------------
// Phase 1: one wave per edge.
//   - all 32 lanes scatter-add the 64-float feature row into agg[dst]
//   - lane 0 inserts key into open-addressed hash + pushes edge on key's list
// ---------------------------------------------------------------------------
__global__ __launch_bounds__(256)
void ec_build_kernel(const float* __restrict__ ef,
                     const int*   __restrict__ esrc,
                     const int*   __restrict__ edst,
                     float*       __restrict__ agg,
                     unsigned long long* __restrict__ hkeys,
                     int*         __restrict__ hhead,
                     int*         __restrict__ hnext,
                     int E, long long N, unsigned int hmask)
{
    const int warp = (blockIdx.x * blockDim.x + threadIdx.x) >> 5;
    const int lane = threadIdx.x & 31;
    if (warp >= E) return;
    const int e = warp;
    const int d = edst[e];

    const float2 f = ((const float2*)(ef + (long long)e * UNITS))[lane];
    float* arow = agg + (long long)d * UNITS + 2 * lane;
    atomicAdd(arow,     f.x);
    atomicAdd(arow + 1, f.y);

    if (lane == 0) {
        const int s = esrc[e];
        const unsigned long long key = (unsigned long long)((long long)s * N + d);
        unsigned int slot = hash64(key) & hmask;
        for (;;) {
            unsigned long long prev = atomicCAS(&hkeys[slot], EMPTY64, key);
            if (prev == EMPTY64 || prev == key) break;
            slot = (slot + 1) & hmask;
        }
        hnext[e] = atomicExch(&hhead[slot], e);
    }
}

// ---------------------------------------------------------------------------
// Phase 2: 1 wave == one 16-edge tile. Build h=[ef | agg[src]-rev] in LDS,
// then 16x16x4 fp32 WMMA over K=128 into 4 column tiles of 16.
// A-fragment layout (ISA 7.12.2, 32-bit A 16x4): lane L -> M=L%16,
//   VGPR v -> K = v + 2*(L/16).  B (4x16): lane L -> N=L%16, VGPR v ->
//   K = v + 2*(L/16) (mirrors A / C-D striping).  C/D: VGPR i -> M=i+8*(L/16),
//   N=L%16.
// ---------------------------------------------------------------------------
__global__ __launch_bounds__(256)
void ec_update_kernel(const float* __restrict__ ef,
                      const int*   __restrict__ esrc,
                      const int*   __restrict__ edst,
                      const float* __restrict__ W,     // [128,64] row-major
                      const float* __restrict__ bias,  // [64]
                      const float* __restrict__ agg,
                      const unsigned long long* __restrict__ hkeys,
                      const int*   __restrict__ hhead,
                      const int*   __restrict__ hnext,
                      float*       __restrict__ out,
                      int E, long long N, unsigned int hmask)
{
    __shared__ float Wt[UNITS][HPAD];      // Wt[n][k] = W[k][n]
    __shared__ float hT[8][16][HPAD];      // per-wave 16x128 tiles (padded)

    const int tid  = threadIdx.x;
    const int lane = tid & 31;
    const int wave = tid >> 5;
    const int half = lane >> 4;            // 0 | 1
    const int l16  = lane & 15;

    // Stage W transposed into LDS (one time per block, coalesced global reads)
    for (int i = tid; i < TWOU * UNITS; i += 256) {
        const int k = i >> 6;              // 0..127
        const int n = i & 63;              // 0..63
        Wt[n][k] = W[i];
    }
    __syncthreads();

    const int numTiles = (E + 15) >> 4;
    const int tile     = blockIdx.x * 8 + wave;   // wave-uniform
    const int ebase    = tile << 4;

    if (tile < numTiles) {
        // ---- lanes 0..15: hash probe for this tile's 16 edges -------------
        int myHead = -1, mySrc = 0, myValid = 0;
        if (lane < 16) {
            const int e = ebase + lane;
            if (e < E) {
                myValid = 1;
                const int s = esrc[e];
                const int d = edst[e];
                mySrc = s;
                const unsigned long long rk =
                    (unsigned long long)((long long)d * N + s);
                unsigned int slot = hash64(rk) & hmask;
                for (;;) {
                    const unsigned long long kk = hkeys[slot];
                    if (kk == EMPTY64) { myHead = -1; break; }
                    if (kk == rk)      { myHead = hhead[slot]; break; }
                    slot = (slot + 1) & hmask;
                }
            }
        }

        // ---- cooperative build of h rows (2 cols per lane) ---------------
        for (int m = 0; m < 16; ++m) {
            const int v = __shfl(myValid, m);
            float2 efv = make_float2(0.f, 0.f);
            float2 msg = make_float2(0.f, 0.f);
            if (v) {
                const int e    = ebase + m;
                const int srcm = __shfl(mySrc, m);
                int       j    = __shfl(myHead, m);
                msg = ((const float2*)(agg + (long long)srcm * UNITS))[lane];
                efv = ((const float2*)(ef  + (long long)e    * UNITS))[lane];
                while (j >= 0) {            // per-key multi-edge list (len ~1)
                    const float2 fj =
                        ((const float2*)(ef + (long long)j * UNITS))[lane];
                    msg.x -= fj.x; msg.y -= fj.y;
                    j = hnext[j];
                }
            }
            float* row = &hT[wave][m][0];
            ((float2*)row)[lane]           = efv;   // cols 0..63
            ((float2*)(row + UNITS))[lane] = msg;   // cols 64..127
        }
    }
    __syncthreads();                        // uniform barrier: LDS visible

    if (tile < numTiles) {
        // ---- WMMA: D(16x64) = h(16x128) @ W(128x64), K-step 4 -------------
        v8f c0 = {}, c1 = {}, c2 = {}, c3 = {};
        const float* hrow = &hT[wave][l16][0];
        const int koff = 2 * half;

        #pragma unroll 4
        for (int kb = 0; kb < TWOU; kb += 4) {
            const v2f a  = *(const v2f*)(hrow + kb + koff);
            const v2f b0 = *(const v2f*)(&Wt[ 0 + l16][kb + koff]);
            const v2f b1 = *(const v2f*)(&Wt[16 + l16][kb + koff]);
            const v2f b2 = *(const v2f*)(&Wt[32 + l16][kb + koff]);
            const v2f b3 = *(const v2f*)(&Wt[48 + l16][kb + koff]);
            c0 = __builtin_amdgcn_wmma_f32_16x16x4_f32(false, a, false, b0,
                                                       (short)0, c0, false, false);
            c1 = __builtin_amdgcn_wmma_f32_16x16x4_f32(false, a, false, b1,
                                                       (short)0, c1, false, false);
            c2 = __builtin_amdgcn_wmma_f32_16x16x4_f32(false, a, false, b2,
                                                       (short)0, c2, false, false);
            c3 = __builtin_amdgcn_wmma_f32_16x16x4_f32(false, a, false, b3,
                                                       (short)0, c3, false, false);
        }

        // ---- epilogue: bias + relu + store --------------------------------
        const float bb0 = bias[ 0 + l16];
        const float bb1 = bias[16 + l16];
        const float bb2 = bias[32 + l16];
        const float bb3 = bias[48 + l16];
        #pragma unroll
        for (int i = 0; i < 8; ++i) {
            const int M = i + 8 * half;     // C/D striping
            const int e = ebase + M;
            if (e < E) {
                float* orow = out + (long long)e * UNITS;
                orow[ 0 + l16] = fmaxf(c0[i] + bb0, 0.f);
                orow[16 + l16] = fmaxf(c1[i] + bb1, 0.f);
                orow[32 + l16] = fmaxf(c2[i] + bb2, 0.f);
                orow[48 + l16] = fmaxf(c3[i] + bb3, 0.f);
            }
        }
    }
}

// ---------------------------------------------------------------------------
static inline size_t alignUp(size_t x, size_t a) { return (x + a - 1) & ~(a - 1); }

extern "C" void kernel_launch(void* const* d_in, const int* in_sizes, int n_in,
                              void* d_out, int out_size, void* d_ws, size_t ws_size,
                              hipStream_t stream) {
    const float* ef   = (const float*)d_in[0];
    const int*   esrc = (const int*)  d_in[1];
    const int*   edst = (const int*)  d_in[2];
    // d_in[3] graph_indicator unused (node ids globally disjoint)
    // d_in[4] num_nodes scalar: host value recovered from graph_indicator size
    const float* W    = (const float*)d_in[5];
    const float* bias = (const float*)d_in[6];
    float*       out  = (float*)d_out;

    const int E         = in_sizes[1];
    const int num_nodes = in_sizes[3];
    const long long N   = (long long)num_nodes;

    unsigned int hsize = 1;
    while (hsize < (unsigned int)(2 * (long long)E)) hsize <<= 1;
    const unsigned int hmask = hsize - 1;

    // Workspace layout
    char* ws = (char*)d_ws;
    size_t off = 0;
    float* agg = (float*)(ws + off);
    off = alignUp(off + (size_t)num_nodes * UNITS * sizeof(float), 256);
    unsigned long long* hkeys = (unsigned long long*)(ws + off);
    off = alignUp(off + (size_t)hsize * sizeof(unsigned long long), 256);
    int* hhead = (int*)(ws + off);
    off = alignUp(off + (size_t)hsize * sizeof(int), 256);
    int* hnext = (int*)(ws + off);
    off = alignUp(off + (size_t)E * sizeof(int), 256);
    (void)ws_size; (void)n_in; (void)out_size;

    hipMemsetAsync(agg,   0,    (size_t)num_nodes * UNITS * sizeof(float), stream);
    hipMemsetAsync(hkeys, 0xFF, (size_t)hsize * sizeof(unsigned long long), stream);
    hipMemsetAsync(hhead, 0xFF, (size_t)hsize * sizeof(int), stream);

    // Phase 1: aggregation + hash build (warp per edge)
    {
        const long long threads = (long long)E * 32;
        const int blocks = (int)((threads + 255) / 256);
        ec_build_kernel<<<blocks, 256, 0, stream>>>(
            ef, esrc, edst, agg, hkeys, hhead, hnext, E, N, hmask);
    }

    // Phase 2: message + WMMA GEMM + relu (wave per 16-edge tile, 8 waves/block)
    {
        const int numTiles = (E + 15) >> 4;
        const int blocks   = (numTiles + 7) / 8;
        ec_update_kernel<<<blocks, 256, 0, stream>>>(
            ef, esrc, edst, W, bias, agg, hkeys, hhead, hnext, out,
            E, N, hmask);
    }
}